// CTCOutputLayer_69037304316393
// MI455X (gfx1250) — compile-verified
//
#include <hip/hip_runtime.h>
#include <math.h>

#define NEGF (-1e30f)

#if __has_builtin(__builtin_amdgcn_s_wait_asynccnt)
#define WAIT_ASYNC(n) __builtin_amdgcn_s_wait_asynccnt(n)
#else
#define WAIT_ASYNC(n) asm volatile("s_wait_asynccnt %0" :: "i"(n) : "memory")
#endif

// Async copy of 16 bytes per lane from global to LDS (CDNA5, ASYNCcnt-tracked).
// ldsAddr: byte offset within the workgroup's LDS allocation
// voff:    per-lane byte offset from the uniform sgpr base
__device__ __forceinline__ void async_load_b128_to_lds(unsigned ldsAddr,
                                                       unsigned voff,
                                                       unsigned long long base) {
  asm volatile("global_load_async_to_lds_b128 %0, %1, %2"
               :: "v"(ldsAddr), "v"(voff), "s"(base)
               : "memory");
}

// ---------------------------------------------------------------------------
// Kernel 0: labels[b,s] = argmax_v targets[b,s,v]   (V=256, one wave per row)
// Per-lane contiguous float4 loads -> 2x global_load_b128 per row.
// ---------------------------------------------------------------------------
__global__ __launch_bounds__(256) void argmax_rows_kernel(
    const float* __restrict__ tt, int* __restrict__ labels, int nrows) {
  int row  = blockIdx.x * 8 + (threadIdx.x >> 5);
  int lane = threadIdx.x & 31;
  if (row >= nrows) return;
  const float4* p = (const float4*)(tt + (size_t)row * 256 + lane * 8);
  float4 q0 = p[0];
  float4 q1 = p[1];
  float v[8] = {q0.x, q0.y, q0.z, q0.w, q1.x, q1.y, q1.z, q1.w};
  float bv = -INFINITY;
  int   bi = 0;
#pragma unroll
  for (int i = 0; i < 8; ++i) {
    if (v[i] > bv) { bv = v[i]; bi = lane * 8 + i; }
  }
#pragma unroll
  for (int m = 16; m; m >>= 1) {
    float ov = __shfl_xor(bv, m, 32);
    int   oi = __shfl_xor(bi, m, 32);
    if (ov > bv || (ov == bv && oi < bi)) { bv = ov; bi = oi; }
  }
  if (lane == 0) labels[row] = bi;
}

// ---------------------------------------------------------------------------
// Kernel 1: logZ[b,t] = logsumexp_v logits[b,t,v]   (V=256, one wave per row)
// ---------------------------------------------------------------------------
__global__ __launch_bounds__(256) void logz_rows_kernel(
    const float* __restrict__ logits, float* __restrict__ logZ, int nrows) {
  int row  = blockIdx.x * 8 + (threadIdx.x >> 5);
  int lane = threadIdx.x & 31;
  if (row >= nrows) return;
  const float4* p = (const float4*)(logits + (size_t)row * 256 + lane * 8);
  float4 q0 = p[0];
  float4 q1 = p[1];
  float v[8] = {q0.x, q0.y, q0.z, q0.w, q1.x, q1.y, q1.z, q1.w};
  float mx = -INFINITY;
#pragma unroll
  for (int i = 0; i < 8; ++i) mx = fmaxf(mx, v[i]);
#pragma unroll
  for (int m = 16; m; m >>= 1) mx = fmaxf(mx, __shfl_xor(mx, m, 32));
  float s = 0.f;
#pragma unroll
  for (int i = 0; i < 8; ++i) s += __expf(v[i] - mx);
#pragma unroll
  for (int m = 16; m; m >>= 1) s += __shfl_xor(s, m, 32);
  if (lane == 0) logZ[row] = mx + __logf(s);
}

// ---------------------------------------------------------------------------
// Kernel 2: CTC alpha recursion, one workgroup per batch element.
//   288 threads (9 wave32), one extended-label state per thread (L2 = 257).
//   8-deep async-prefetch ring of logits rows (1KB each) in LDS; each row
//   fetched by lanes tid<64 with global_load_async_to_lds_b128 (16B/lane).
// ---------------------------------------------------------------------------
__global__ __launch_bounds__(288) void ctc_alpha_kernel(
    const float* __restrict__ logits, const float* __restrict__ logZ,
    const int* __restrict__ labels, const int* __restrict__ lengths,
    float* __restrict__ nll_out, int T, int S) {
  const int V  = 256;
  const int L2 = 2 * S + 1;            // 257 for S=128
  const int b   = blockIdx.x;
  const int tid = threadIdx.x;
  const int s   = tid;

  __shared__ __align__(16) float rowbuf[8 * 256];  // prefetch ring, 8KB
  __shared__ float zrow[1024];         // logZ[b, :], 4KB  (T <= 1024)
  __shared__ float A[2][2 + 257];      // alpha double buffer, NEG-padded
  __shared__ int   extS[257];

  if (tid < 2) { A[0][tid] = NEGF; A[1][tid] = NEGF; }
  for (int i = tid; i < T; i += 288) zrow[i] = logZ[(size_t)b * T + i];
  if (s < L2) extS[s] = (s & 1) ? labels[b * S + (s >> 1)] : 0;

  const unsigned long long base = (unsigned long long)(size_t)(logits + (size_t)b * T * V);
  const unsigned ldsRowBase = (unsigned)(size_t)&rowbuf[0];

  // Prologue: issue rows 0..7 into the ring (lanes 0..63, 16B each).
  if (tid < 64) {
#pragma unroll
    for (int r = 0; r < 8; ++r)
      async_load_b128_to_lds(ldsRowBase + (unsigned)(r * 1024 + tid * 16),
                             (unsigned)(r * 1024 + tid * 16), base);
  }
  WAIT_ASYNC(7);                       // row 0 complete (in-order)
  __syncthreads();                     // extS / zrow / pads / row 0 visible

  int  myExt = 0;
  bool skip  = false;
  if (s < L2) {
    myExt = extS[s];
    skip  = (s >= 2) && (myExt != 0) && (myExt != extS[s - 2]);
  }
  // t = 0 init: alpha0[0] = lp(blank), alpha0[1] = lp(label0), else NEG.
  if (s < L2) A[0][2 + s] = (s < 2) ? (rowbuf[myExt] - zrow[0]) : NEGF;

  int cur = 0;
  for (int t = 1; t < T; ++t) {
    WAIT_ASYNC(6);                     // row t complete (<=6 newer outstanding)
    __syncthreads();                   // publish row t + alpha[cur]
    // Issue row t+7 (clamped at the tail so exactly one issue per iteration;
    // redundant tail loads land in a slot that is never read again).
    if (tid < 64) {
      int rf = t + 7; if (rf > T - 1) rf = T - 1;
      async_load_b128_to_lds(ldsRowBase + (unsigned)(((t + 7) & 7) * 1024 + tid * 16),
                             (unsigned)((unsigned)rf * 1024 + tid * 16), base);
    }
    if (s < L2) {
      const float* Ac = A[cur];
      float a  = Ac[2 + s];
      float a1 = Ac[1 + s];
      float a2 = skip ? Ac[s] : NEGF;
      float m  = fmaxf(a, fmaxf(a1, a2));
      float sum = __expf(a - m) + __expf(a1 - m) + __expf(a2 - m);
      float lp  = rowbuf[(t & 7) * 256 + myExt] - zrow[t];
      A[cur ^ 1][2 + s] = m + __logf(sum) + lp;
    }
    cur ^= 1;
  }
  __syncthreads();
  if (tid == 0) {
    int len = lengths[b];
    float x = A[cur][2 + (2 * len - 1)];
    float y = A[cur][2 + (2 * len)];
    float m = fmaxf(x, y);
    float nll = -(m + __logf(__expf(x - m) + __expf(y - m)));
    nll_out[b] = nll / (float)len;
  }
}

// ---------------------------------------------------------------------------
// Kernel 3: out = mean_b nll_out[b]
// ---------------------------------------------------------------------------
__global__ __launch_bounds__(256) void finalize_kernel(
    const float* __restrict__ nll, float* __restrict__ out, int B) {
  __shared__ float sh[256];
  int t = threadIdx.x;
  sh[t] = (t < B) ? nll[t] : 0.f;
  __syncthreads();
  for (int off = 128; off; off >>= 1) {
    if (t < off) sh[t] += sh[t + off];
    __syncthreads();
  }
  if (t == 0) out[0] = sh[0] / (float)B;
}

extern "C" void kernel_launch(void* const* d_in, const int* in_sizes, int n_in,
                              void* d_out, int out_size, void* d_ws, size_t ws_size,
                              hipStream_t stream) {
  const float* logits  = (const float*)d_in[0];  // [B, T, V] fp32
  const float* targets = (const float*)d_in[1];  // [B, S, V] fp32
  const int*   lengths = (const int*)d_in[2];    // [B] int32

  const int B = in_sizes[2];
  const int V = 256;
  const int T = in_sizes[0] / (B * V);           // 1024
  const int S = in_sizes[1] / (B * V);           // 128

  // Workspace layout (256B-aligned slices): labels[B*S] | logZ[B*T] | nll[B]
  char*  ws     = (char*)d_ws;
  size_t off0   = 0;
  int*   labels = (int*)(ws + off0);
  size_t off1   = (off0 + (size_t)B * S * sizeof(int) + 255) & ~(size_t)255;
  float* logZ   = (float*)(ws + off1);
  size_t off2   = (off1 + (size_t)B * T * sizeof(float) + 255) & ~(size_t)255;
  float* nll    = (float*)(ws + off2);

  {
    int rows = B * S;
    argmax_rows_kernel<<<(rows + 7) / 8, 256, 0, stream>>>(targets, labels, rows);
  }
  {
    int rows = B * T;
    logz_rows_kernel<<<(rows + 7) / 8, 256, 0, stream>>>(logits, logZ, rows);
  }
  ctc_alpha_kernel<<<B, 288, 0, stream>>>(logits, logZ, labels, lengths, nll, T, S);
  finalize_kernel<<<1, 256, 0, stream>>>(nll, (float*)d_out, B);
}